// ProposalGeneration_10307921511132
// MI455X (gfx1250) — compile-verified
//
#include <hip/hip_runtime.h>
#include <hip/hip_bf16.h>

typedef float v2f __attribute__((ext_vector_type(2)));
typedef float v8f __attribute__((ext_vector_type(8)));

#define T_  128
#define L_  32
#define C_  4
#define B_  32
#define D_  512
#define RM  4096            // L*L*C flattened rows of Wc
#define TM  64              // M tile per block
#define TN  128             // N tile per block
#define A_STRIDE 140        // padded LDS row stride (dwords): bank-stride 12 -> conflict free b64 frags
#define B_STRIDE 136        // padded LDS row stride (dwords): 544B rows (16B aligned), conflict-free b32 frags

__global__ __launch_bounds__(256)
void pg_gemm_kernel(const float* __restrict__ f,
                    const float* __restrict__ mask,
                    const float* __restrict__ Wc,
                    float* __restrict__ fc,
                    float* __restrict__ fm)
{
    __shared__ float As[TM * A_STRIDE];   // 35840 B
    __shared__ float Bs[T_ * B_STRIDE];   // 69632 B

    const int nTile = blockIdx.x;   // 0..3
    const int mTile = blockIdx.y;   // 0..63
    const int b     = blockIdx.z;   // 0..31

    const int tid   = threadIdx.x;
    const int lane  = tid & 31;
    const int wave  = tid >> 5;     // 0..7
    const int waveM = wave & 3;     // 4 waves along M (16 rows each)
    const int waveN = wave >> 2;    // 2 waves along N (64 cols each)

    // ---- stage A: Wc rows [mTile*64, +64), all K=128 (contiguous region) ----
    {
        const float4* src = (const float4*)(Wc + (size_t)mTile * TM * T_);
        #pragma unroll
        for (int i = 0; i < 8; ++i) {
            int idx  = tid + i * 256;         // 0..2047 float4s
            int row  = idx >> 5;              // 32 float4 per 128-float row
            int col4 = idx & 31;
            float4 v = src[idx];
            *(float4*)&As[row * A_STRIDE + col4 * 4] = v;
        }
    }
    // ---- stage B: f[b][0..127][nTile*128 .. +128) ----
    {
        const float* srcb = f + (size_t)b * T_ * D_ + nTile * TN;
        #pragma unroll
        for (int i = 0; i < 16; ++i) {
            int idx  = tid + i * 256;         // 0..4095 float4s
            int row  = idx >> 5;              // 32 float4 per 128-float row
            int col4 = idx & 31;
            float4 v = *(const float4*)(srcb + (size_t)row * D_ + col4 * 4);
            *(float4*)&Bs[row * B_STRIDE + col4 * 4] = v;
        }
    }
    __syncthreads();

    // ---- WMMA main loop: D(16x16) += A(16x4) * B(4x16), K = 128 in steps of 4 ----
    v8f acc[4] = {v8f{}, v8f{}, v8f{}, v8f{}};

    const int ml      = lane & 15;                 // A row within 16 / B col within 16
    const int halfOff = (lane < 16) ? 0 : 2;       // K offset for upper lane half
    const int n0      = waveN * 64;                // wave's N base within tile
    const float* Arow = &As[(waveM * 16 + ml) * A_STRIDE];

    for (int k = 0; k < T_; k += 4) {
        // A fragment: lane holds A[m=ml][k+halfOff], A[m=ml][k+halfOff+1]  (b64 ds load)
        v2f a = *(const v2f*)(Arow + k + halfOff);

        const float* Bk0 = &Bs[(k + halfOff    ) * B_STRIDE + n0 + ml];
        const float* Bk1 = &Bs[(k + halfOff + 1) * B_STRIDE + n0 + ml];
        v2f b0 = { Bk0[ 0], Bk1[ 0] };
        v2f b1 = { Bk0[16], Bk1[16] };
        v2f b2 = { Bk0[32], Bk1[32] };
        v2f b3 = { Bk0[48], Bk1[48] };

        acc[0] = __builtin_amdgcn_wmma_f32_16x16x4_f32(false, a, false, b0, (short)0, acc[0], false, false);
        acc[1] = __builtin_amdgcn_wmma_f32_16x16x4_f32(false, a, false, b1, (short)0, acc[1], false, false);
        acc[2] = __builtin_amdgcn_wmma_f32_16x16x4_f32(false, a, false, b2, (short)0, acc[2], false, false);
        acc[3] = __builtin_amdgcn_wmma_f32_16x16x4_f32(false, a, false, b3, (short)0, acc[3], false, false);
    }

    // ---- epilogue: mask, store fc, fold c-rows for fm ----
    const int rowBase = mTile * TM + waveM * 16;             // multiple of 16
    const int gBase   = (rowBase >> 2) + ((lane < 16) ? 0 : 2); // (l,m) group for acc rows v=0..3
    const float mLo = mask[b * (L_ * L_) + gBase];           // rows v=0..3 of this lane-half
    const float mHi = mask[b * (L_ * L_) + gBase + 1];       // rows v=4..7

    #pragma unroll
    for (int j = 0; j < 4; ++j) {
        #pragma unroll
        for (int v = 0; v < 4; ++v) acc[j][v]     *= mLo;
        #pragma unroll
        for (int v = 0; v < 4; ++v) acc[j][v + 4] *= mHi;
    }

    const int dBase = nTile * TN + n0 + ml;                  // + j*16 per sub-tile
    float* fcb = fc + (size_t)b * RM * D_;
    float* fmb = fm + (size_t)b * (L_ * L_) * D_;
    const int rowHalf = rowBase + ((lane < 16) ? 0 : 8);

    #pragma unroll
    for (int j = 0; j < 4; ++j) {
        const int d = dBase + j * 16;
        #pragma unroll
        for (int v = 0; v < 8; ++v) {
            fcb[(size_t)(rowHalf + v) * D_ + d] = acc[j][v];
        }
        // fm: mean over c (4 consecutive rows), mask already applied
        float slo = (acc[j][0] + acc[j][1] + acc[j][2] + acc[j][3]) * 0.25f;
        float shi = (acc[j][4] + acc[j][5] + acc[j][6] + acc[j][7]) * 0.25f;
        fmb[(size_t)gBase       * D_ + d] = slo;
        fmb[(size_t)(gBase + 1) * D_ + d] = shi;
    }
}

// fb[b,l,d] = mean over the 4 frames of segment l  (T/L == 4)
__global__ __launch_bounds__(256)
void pg_fb_kernel(const float* __restrict__ f, float* __restrict__ fbout)
{
    int idx = blockIdx.x * 256 + threadIdx.x;    // over B*L*D = 524288
    int d   = idx & (D_ - 1);
    int bl  = idx >> 9;                          // b*L + l
    const float* p = f + (size_t)bl * 4 * D_ + d;
    fbout[idx] = (p[0] + p[D_] + p[2 * D_] + p[3 * D_]) * 0.25f;
}

extern "C" void kernel_launch(void* const* d_in, const int* in_sizes, int n_in,
                              void* d_out, int out_size, void* d_ws, size_t ws_size,
                              hipStream_t stream) {
    const float* f    = (const float*)d_in[0];   // [B, T, D]
    const float* mask = (const float*)d_in[1];   // [B, L, L]
    const float* Wc   = (const float*)d_in[2];   // [L, L, C, T] == [RM, T_]

    float* fc = (float*)d_out;                                  // [B, RM, D]
    float* fm = fc + (size_t)B_ * RM * D_;                      // [B, L*L, D]
    float* fb = fm + (size_t)B_ * L_ * L_ * D_;                 // [B, L, D]

    dim3 grid(TN == 128 ? D_ / TN : 1, RM / TM, B_);
    pg_gemm_kernel<<<grid, 256, 0, stream>>>(f, mask, Wc, fc, fm);

    pg_fb_kernel<<<(B_ * L_ * D_) / 256, 256, 0, stream>>>(f, fb);
}